// BaseGAT_60318520705767
// MI455X (gfx1250) — compile-verified
//
#include <hip/hip_runtime.h>

// ---------------------------------------------------------------------------
// GATv2 2-layer forward for MI455X (gfx1250, wave32).
// Dense transforms via V_WMMA_F32_16X16X4_F32 (full fp32 precision),
// 16x32 output tile per wave = two independent accumulator chains to hide
// the WMMA->WMMA RAW hazard and halve A-fragment traffic.
// Edge softmax/aggregation via wave-per-edge gather + float atomics.
// ---------------------------------------------------------------------------

#define NN    100000
#define NE    500000
#define F_IN  128
#define HID   64
#define HEADS 4
#define D1    256        // HEADS*HID
#define OUTC  32
#define SLOPE 0.2f

typedef __attribute__((ext_vector_type(2))) float v2f;
typedef __attribute__((ext_vector_type(8))) float v8f;

// ----- order-preserving float<->uint encoding for atomicMax-based segment max
__device__ __forceinline__ unsigned encOrd(float f) {
  unsigned u = __float_as_uint(f);
  return (u & 0x80000000u) ? ~u : (u | 0x80000000u);
}
__device__ __forceinline__ float decOrd(unsigned u) {
  return (u & 0x80000000u) ? __uint_as_float(u ^ 0x80000000u)
                           : __uint_as_float(~u);
}
#define ENC_NEG_INF 0x007FFFFFu   // encOrd(-inf)

// ---------------------------------------------------------------------------
// Y[rows x cols] = X[rows x K] @ W[K x cols] + bias   (fp32 WMMA, K%4==0,
// rows%16==0, cols%32==0).  One wave per 16x32 output tile (2 accumulators).
// A frag (16x4): lane L -> row = L%16, K-pair = (L/16)*2.
// B frag (4x16): lane L -> col = L%16, K-pair = (L/16)*2.
// C/D (16x16):   lane L, vgpr v -> row = v + 8*(L/16), col = L%16.
// ---------------------------------------------------------------------------
__global__ void gemm_wmma_f32(const float* __restrict__ X,
                              const float* __restrict__ W,
                              const float* __restrict__ bias,
                              float* __restrict__ Y,
                              int rows, int K, int cols) {
  const int wavesPerBlock = blockDim.x >> 5;
  const int wave = blockIdx.x * wavesPerBlock + (threadIdx.x >> 5);
  const int lane = threadIdx.x & 31;
  const int tilesN2 = cols >> 5;                 // 32-wide column tile pairs
  const int totalTiles = (rows >> 4) * tilesN2;
  if (wave >= totalTiles) return;                // wave-uniform: EXEC all-1

  const int tm = wave / tilesN2;
  const int tn = wave - tm * tilesN2;
  const int r     = lane & 15;                   // A row / B,D col within tile
  const int khalf = (lane >> 4) << 1;            // 0 or 2

  const float* xrow = X + (size_t)(tm * 16 + r) * K;
  const int    nc0  = tn * 32 + r;               // first 16-col tile
  const int    nc1  = nc0 + 16;                  // second 16-col tile

  v8f c0 = {}, c1 = {};
  for (int k = 0; k < K; k += 4) {
    v2f a, b0, b1;
    const float* wrow0 = W + (size_t)(k + khalf)     * cols;
    const float* wrow1 = W + (size_t)(k + khalf + 1) * cols;
    a.x  = xrow[k + khalf];
    a.y  = xrow[k + khalf + 1];
    b0.x = wrow0[nc0];  b0.y = wrow1[nc0];
    b1.x = wrow0[nc1];  b1.y = wrow1[nc1];
    c0 = __builtin_amdgcn_wmma_f32_16x16x4_f32(false, a, false, b0,
                                               (short)0, c0, false, false);
    c1 = __builtin_amdgcn_wmma_f32_16x16x4_f32(false, a, false, b1,
                                               (short)0, c1, false, false);
  }

  const float bv0   = bias[nc0];
  const float bv1   = bias[nc1];
  const int   mbase = tm * 16 + ((lane >> 4) << 3);
#pragma unroll
  for (int v = 0; v < 8; ++v) {
    float* yrow = Y + (size_t)(mbase + v) * cols;
    yrow[nc0] = c0[v] + bv0;
    yrow[nc1] = c1[v] + bv1;
  }
}

// ---------------------------------------------------------------------------
// init helpers
// ---------------------------------------------------------------------------
__global__ void fill_f32(float* p, float v, int n) {
  int t = blockIdx.x * blockDim.x + threadIdx.x;
  if (t < n) p[t] = v;
}
__global__ void fill_u32(unsigned* p, unsigned v, int n) {
  int t = blockIdx.x * blockDim.x + threadIdx.x;
  if (t < n) p[t] = v;
}

// ---------------------------------------------------------------------------
// Layer-1 edge kernels (H=4, C=64): wave per edge, 8 channels per lane.
// ---------------------------------------------------------------------------
__global__ void edge_alpha_l1(const float* __restrict__ xl,
                              const float* __restrict__ xr,
                              const float* __restrict__ eattr,
                              const float* __restrict__ We,   // [D1]
                              const float* __restrict__ att,  // [HEADS*HID]
                              const int* __restrict__ src,
                              const int* __restrict__ dst,
                              float* __restrict__ alpha,      // [E*HEADS]
                              unsigned* __restrict__ amax,    // [N*HEADS]
                              int nE) {
  const int e = blockIdx.x * (blockDim.x >> 5) + (threadIdx.x >> 5);
  if (e >= nE) return;
  const int lane = threadIdx.x & 31;
  const int s = src[e], d = dst[e];
  const float ea = eattr[e];

  const float* pl = xl  + (size_t)s * D1 + lane * 8;
  const float* pr = xr  + (size_t)d * D1 + lane * 8;
  const float* pw = We  + lane * 8;
  const float* pa = att + lane * 8;

  float partial = 0.f;
#pragma unroll
  for (int i = 0; i < 8; ++i) {
    float v = pl[i] + pr[i] + ea * pw[i];
    v = (v > 0.f) ? v : v * SLOPE;
    partial += v * pa[i];
  }
  // reduce across the 8 lanes of this head
  partial += __shfl_down(partial, 4, 8);
  partial += __shfl_down(partial, 2, 8);
  partial += __shfl_down(partial, 1, 8);
  if ((lane & 7) == 0) {
    const int h = lane >> 3;
    alpha[(size_t)e * HEADS + h] = partial;
    atomicMax(&amax[(size_t)d * HEADS + h], encOrd(partial));
  }
}

__global__ void edge_exp(const float* __restrict__ alpha_in,
                         float* __restrict__ ealpha,
                         const unsigned* __restrict__ amax,
                         const int* __restrict__ dst,
                         float* __restrict__ denom,
                         int nEH, int heads) {
  const int t = blockIdx.x * blockDim.x + threadIdx.x;
  if (t >= nEH) return;
  const int e = t / heads, h = t - e * heads;
  const int d = dst[e];
  const float m  = decOrd(amax[(size_t)d * heads + h]);
  const float ex = expf(alpha_in[t] - m);
  ealpha[t] = ex;
  atomicAdd(&denom[(size_t)d * heads + h], ex);
}

__global__ void edge_aggr_l1(const float* __restrict__ xl,
                             const float* __restrict__ ealpha,
                             const float* __restrict__ denom,
                             const int* __restrict__ src,
                             const int* __restrict__ dst,
                             float* __restrict__ acc,  // [N*D1], pre-zeroed
                             int nE) {
  const int e = blockIdx.x * (blockDim.x >> 5) + (threadIdx.x >> 5);
  if (e >= nE) return;
  const int lane = threadIdx.x & 31;
  const int s = src[e], d = dst[e];
  const int h = lane >> 3;
  const float w = ealpha[(size_t)e * HEADS + h] /
                  (denom[(size_t)d * HEADS + h] + 1e-16f);
  const float* pl = xl  + (size_t)s * D1 + lane * 8;
  float*       po = acc + (size_t)d * D1 + lane * 8;
#pragma unroll
  for (int i = 0; i < 8; ++i)
    atomicAdd(&po[i], pl[i] * w);
}

__global__ void bias_relu_inplace(float* __restrict__ A,
                                  const float* __restrict__ bias,
                                  int total, int cols) {
  const int t = blockIdx.x * blockDim.x + threadIdx.x;
  if (t >= total) return;
  const float v = A[t] + bias[t % cols];
  A[t] = (v > 0.f) ? v : 0.f;
}

// ---------------------------------------------------------------------------
// Layer-2 edge kernels (H=1, C=32): wave per edge, 1 channel per lane.
// ---------------------------------------------------------------------------
__global__ void edge_alpha_l2(const float* __restrict__ xl,
                              const float* __restrict__ xr,
                              const float* __restrict__ eattr,
                              const float* __restrict__ We,   // [OUTC]
                              const float* __restrict__ att,  // [OUTC]
                              const int* __restrict__ src,
                              const int* __restrict__ dst,
                              float* __restrict__ alpha,      // [E]
                              unsigned* __restrict__ amax,    // [N]
                              int nE) {
  const int e = blockIdx.x * (blockDim.x >> 5) + (threadIdx.x >> 5);
  if (e >= nE) return;
  const int lane = threadIdx.x & 31;
  const int s = src[e], d = dst[e];
  const float ea = eattr[e];
  float v = xl[(size_t)s * OUTC + lane] + xr[(size_t)d * OUTC + lane] +
            ea * We[lane];
  v = (v > 0.f) ? v : v * SLOPE;
  float partial = v * att[lane];
  partial += __shfl_down(partial, 16);
  partial += __shfl_down(partial, 8);
  partial += __shfl_down(partial, 4);
  partial += __shfl_down(partial, 2);
  partial += __shfl_down(partial, 1);
  if (lane == 0) {
    alpha[e] = partial;
    atomicMax(&amax[d], encOrd(partial));
  }
}

__global__ void edge_aggr_l2(const float* __restrict__ xl,
                             const float* __restrict__ ealpha,
                             const float* __restrict__ denom,
                             const int* __restrict__ src,
                             const int* __restrict__ dst,
                             float* __restrict__ acc,  // [N*OUTC], pre-zeroed
                             int nE) {
  const int e = blockIdx.x * (blockDim.x >> 5) + (threadIdx.x >> 5);
  if (e >= nE) return;
  const int lane = threadIdx.x & 31;
  const int s = src[e], d = dst[e];
  const float w = ealpha[e] / (denom[d] + 1e-16f);
  atomicAdd(&acc[(size_t)d * OUTC + lane],
            xl[(size_t)s * OUTC + lane] * w);
}

__global__ void bias_inplace(float* __restrict__ A,
                             const float* __restrict__ bias,
                             int total, int cols) {
  const int t = blockIdx.x * blockDim.x + threadIdx.x;
  if (t >= total) return;
  A[t] = A[t] + bias[t % cols];
}

// ---------------------------------------------------------------------------
// launch
// ---------------------------------------------------------------------------
extern "C" void kernel_launch(void* const* d_in, const int* in_sizes, int n_in,
                              void* d_out, int out_size, void* d_ws, size_t ws_size,
                              hipStream_t stream) {
  const float* x     = (const float*)d_in[0];
  const int*   eidx  = (const int*)d_in[1];      // jax int64 -> int32 (x64 off)
  const float* eattr = (const float*)d_in[2];
  const float* Wl1   = (const float*)d_in[3];
  const float* bl1   = (const float*)d_in[4];
  const float* Wr1   = (const float*)d_in[5];
  const float* br1   = (const float*)d_in[6];
  const float* We1   = (const float*)d_in[7];
  const float* att1  = (const float*)d_in[8];
  const float* b1    = (const float*)d_in[9];
  const float* Wl2   = (const float*)d_in[10];
  const float* bl2   = (const float*)d_in[11];
  const float* Wr2   = (const float*)d_in[12];
  const float* br2   = (const float*)d_in[13];
  const float* We2   = (const float*)d_in[14];
  const float* att2  = (const float*)d_in[15];
  const float* b2    = (const float*)d_in[16];

  const int* src = eidx;
  const int* dst = eidx + NE;

  // ---- workspace layout (floats), layer-2 buffers alias dead layer-1 ones
  char* ws = (char*)d_ws;
  float*    xl1   = (float*)(ws);                                 // N*D1
  float*    xr1   = (float*)(ws + (size_t)NN * D1 * 4);           // N*D1
  float*    acc1  = (float*)(ws + (size_t)NN * D1 * 8);           // N*D1 (-> h)
  float*    alph  = (float*)(ws + (size_t)NN * D1 * 12);          // E*HEADS
  unsigned* amax  = (unsigned*)((char*)alph + (size_t)NE * HEADS * 4); // N*HEADS
  float*    denom = (float*)((char*)amax + (size_t)NN * HEADS * 4);    // N*HEADS
  float*    xl2   = xl1;                                          // N*OUTC
  float*    xr2   = xr1;                                          // N*OUTC
  float*    out   = (float*)d_out;                                // N*OUTC

  const int BT = 256;                  // threads per block (8 waves)
  const int WPB = BT / 32;
  auto blocksT = [&](int n) { return (n + BT - 1) / BT; };
  auto blocksW = [&](int nWaves) { return (nWaves + WPB - 1) / WPB; };

  // ================= Layer 1 =================
  {
    const int tiles = (NN / 16) * (D1 / 32);     // 16x32 tiles per wave
    gemm_wmma_f32<<<blocksW(tiles), BT, 0, stream>>>(x, Wl1, bl1, xl1, NN, F_IN, D1);
    gemm_wmma_f32<<<blocksW(tiles), BT, 0, stream>>>(x, Wr1, br1, xr1, NN, F_IN, D1);
  }
  fill_f32<<<blocksT(NN * D1), BT, 0, stream>>>(acc1, 0.f, NN * D1);
  fill_u32<<<blocksT(NN * HEADS), BT, 0, stream>>>(amax, ENC_NEG_INF, NN * HEADS);
  fill_f32<<<blocksT(NN * HEADS), BT, 0, stream>>>(denom, 0.f, NN * HEADS);

  edge_alpha_l1<<<blocksW(NE), BT, 0, stream>>>(xl1, xr1, eattr, We1, att1,
                                                src, dst, alph, amax, NE);
  edge_exp<<<blocksT(NE * HEADS), BT, 0, stream>>>(alph, alph, amax, dst, denom,
                                                   NE * HEADS, HEADS);
  edge_aggr_l1<<<blocksW(NE), BT, 0, stream>>>(xl1, alph, denom, src, dst, acc1, NE);
  bias_relu_inplace<<<blocksT(NN * D1), BT, 0, stream>>>(acc1, b1, NN * D1, D1);
  // acc1 now holds h = relu(conv1(x)); xl1/xr1 are dead -> reused below.

  // ================= Layer 2 =================
  {
    const int tiles = (NN / 16) * (OUTC / 32);   // one 16x32 tile per row block
    gemm_wmma_f32<<<blocksW(tiles), BT, 0, stream>>>(acc1, Wl2, bl2, xl2, NN, D1, OUTC);
    gemm_wmma_f32<<<blocksW(tiles), BT, 0, stream>>>(acc1, Wr2, br2, xr2, NN, D1, OUTC);
  }
  fill_f32<<<blocksT(NN * OUTC), BT, 0, stream>>>(out, 0.f, NN * OUTC);
  fill_u32<<<blocksT(NN), BT, 0, stream>>>(amax, ENC_NEG_INF, NN);
  fill_f32<<<blocksT(NN), BT, 0, stream>>>(denom, 0.f, NN);

  edge_alpha_l2<<<blocksW(NE), BT, 0, stream>>>(xl2, xr2, eattr, We2, att2,
                                                src, dst, alph, amax, NE);
  edge_exp<<<blocksT(NE), BT, 0, stream>>>(alph, alph, amax, dst, denom, NE, 1);
  edge_aggr_l2<<<blocksW(NE), BT, 0, stream>>>(xl2, alph, denom, src, dst, out, NE);
  bias_inplace<<<blocksT(NN * OUTC), BT, 0, stream>>>(out, b2, NN * OUTC, OUTC);
}